// ReactionClassificationHead_26723286516089
// MI455X (gfx1250) — compile-verified
//
#include <hip/hip_runtime.h>
#include <hip/hip_bf16.h>
#include <stdint.h>

// Segmented-sum pooling for reaction graphs (sorted batch_ids, random mol_idx).
// Roofline: 1.02 GB read @ 23.3 TB/s => ~44us floor. Streaming reduce with
// register accumulators; atomics only at graph boundaries; ids staged into LDS
// via CDNA5 async global->LDS DMA (ASYNCcnt).

#define CHUNK 128          // rows per wave
#define WAVES_PER_BLOCK 8  // 256 threads
#define ROWS_PER_BLOCK (CHUNK * WAVES_PER_BLOCK)

typedef float  v4f __attribute__((ext_vector_type(4)));
typedef int    v2i __attribute__((ext_vector_type(2)));

__global__ void rxn_zero_kernel(v4f* __restrict__ out, int n4) {
    int i = blockIdx.x * blockDim.x + threadIdx.x;
    if (i < n4) out[i] = (v4f){0.f, 0.f, 0.f, 0.f};
}

__device__ __forceinline__ void flush_acc(float* __restrict__ out, int bid, int lane,
                                          const v4f& a0, const v4f& a1) {
    // out layout: [B, 2*D] = per graph: [reactant 128 | product 128]
    float* p = out + (size_t)bid * 256 + lane * 4;
    unsafeAtomicAdd(p + 0,   a0.x);
    unsafeAtomicAdd(p + 1,   a0.y);
    unsafeAtomicAdd(p + 2,   a0.z);
    unsafeAtomicAdd(p + 3,   a0.w);
    unsafeAtomicAdd(p + 128, a1.x);
    unsafeAtomicAdd(p + 129, a1.y);
    unsafeAtomicAdd(p + 130, a1.z);
    unsafeAtomicAdd(p + 131, a1.w);
}

__global__ void __launch_bounds__(256)
rxn_pool_kernel(const float* __restrict__ node_rep,
                const int*   __restrict__ batch_ids,
                const int*   __restrict__ mol_idx,
                float*       __restrict__ out,
                int N) {
    extern __shared__ int smem[];  // per wave: CHUNK interleaved (bid, mol) pairs

    const int lane = threadIdx.x & 31;
    const int wave = threadIdx.x >> 5;
    const long long base = (long long)(blockIdx.x * WAVES_PER_BLOCK + wave) * CHUNK;
    if (base >= N) return;  // wave-uniform

    // LDS byte offset of this wave's id staging buffer. Low 32 bits of a
    // generic pointer to LDS == workgroup-relative LDS byte address.
    const uint32_t ldsBase = (uint32_t)(uintptr_t)smem + (uint32_t)wave * (CHUNK * 8u);
    const unsigned long long bidsBase = (unsigned long long)(uintptr_t)batch_ids;
    const unsigned long long molsBase = (unsigned long long)(uintptr_t)mol_idx;

    // --- CDNA5 async DMA: stage (batch_id, mol_idx) pairs into LDS ---
    // Interleaved layout: slot i holds {bid[i], mol[i]} so the consume loop
    // needs a single ds_load_b64 per row. Per-lane b32 copies in GVS mode
    // (SGPR base + 32-bit VGPR byte offset); tail addresses clamped in-bounds.
#pragma unroll
    for (int k = 0; k < CHUNK / 32; ++k) {
        int i = k * 32 + lane;
        long long row = base + i;
        if (row >= N) row = N - 1;  // clamped; those slots are never consumed
        uint32_t goff = (uint32_t)row * 4u;
        uint32_t l0 = ldsBase + (uint32_t)i * 8u;       // bid slot
        uint32_t l1 = ldsBase + (uint32_t)i * 8u + 4u;  // mol slot
        asm volatile("global_load_async_to_lds_b32 %0, %1, %2"
                     :: "v"(l0), "v"(goff), "s"(bidsBase) : "memory");
        asm volatile("global_load_async_to_lds_b32 %0, %1, %2"
                     :: "v"(l1), "v"(goff), "s"(molsBase) : "memory");
    }
    asm volatile("s_wait_asynccnt 0" ::: "memory");

    const v2i* __restrict__ ids = (const v2i*)(smem + wave * (CHUNK * 2));

    // Register accumulators: lane owns columns [4*lane, 4*lane+4).
    v4f a0 = (v4f){0.f, 0.f, 0.f, 0.f};  // mol == 0 (reactant)
    v4f a1 = (v4f){0.f, 0.f, 0.f, 0.f};  // mol == 1 (product)
    int cur = ids[0].x;

    const int nrows = (int)((base + CHUNK <= (long long)N) ? CHUNK : (N - base));
    const v4f* __restrict__ src = (const v4f*)node_rep + base * 32 + lane;

#pragma unroll 4
    for (int i = 0; i < nrows; ++i) {
        const v2i bm = ids[i];  // one broadcast ds_load_b64 (bid, mol)
        if (bm.x != cur) {      // wave-uniform branch, ~1 per 488 rows
            flush_acc(out, cur, lane, a0, a1);
            a0 = (v4f){0.f, 0.f, 0.f, 0.f};
            a1 = (v4f){0.f, 0.f, 0.f, 0.f};
            cur = bm.x;
        }
        // Non-temporal b128: 1 GB stream read once -> don't thrash L2
        // (keeps the 4 MB output + id arrays resident across graph replays).
        const v4f v = __builtin_nontemporal_load(&src[(size_t)i * 32]);
        if (bm.y) {
            a1 += v;
        } else {
            a0 += v;
        }
    }
    flush_acc(out, cur, lane, a0, a1);
}

extern "C" void kernel_launch(void* const* d_in, const int* in_sizes, int n_in,
                              void* d_out, int out_size, void* d_ws, size_t ws_size,
                              hipStream_t stream) {
    const float* node_rep  = (const float*)d_in[0];
    const int*   batch_ids = (const int*)d_in[1];
    const int*   mol_idx   = (const int*)d_in[2];
    float*       out       = (float*)d_out;

    const int N = in_sizes[1];  // number of nodes

    // 1) Zero the output (atomics accumulate into it; harness replays the graph).
    {
        int n4 = out_size / 4;
        int blocks = (n4 + 255) / 256;
        rxn_zero_kernel<<<blocks, 256, 0, stream>>>((v4f*)out, n4);
    }

    // 2) Streaming segmented reduce.
    {
        int blocks = (int)((N + ROWS_PER_BLOCK - 1) / ROWS_PER_BLOCK);
        size_t lds_bytes = (size_t)WAVES_PER_BLOCK * CHUNK * 2 * sizeof(int);
        rxn_pool_kernel<<<blocks, WAVES_PER_BLOCK * 32, lds_bytes, stream>>>(
            node_rep, batch_ids, mol_idx, out, N);
    }
}